// ControlNet_20624432955404
// MI455X (gfx1250) — compile-verified
//
#include <hip/hip_runtime.h>

// ControlNet sparse-band linear for MI455X (gfx1250, wave32).
// out[b, i, :] = bias[i*128 : i*128+128]  plus, for output cols 80..127,
// the exact contribution of the 48x32 nonzero weight band of row-block i,
// computed with chained V_WMMA_F32_16X16X4_F32 (bit-exact fp32 MMA).

typedef __attribute__((ext_vector_type(2))) float v2f;
typedef __attribute__((ext_vector_type(8))) float v8f;

#define NM 16384   // 128*128
#define MC 128     // columns per row-block

__global__ __launch_bounds__(32) void controlnet_band_wmma(
    const float* __restrict__ x,     // [64, 16384] fp32
    const float* __restrict__ W,     // [16384, 16384] fp32 (band used)
    const float* __restrict__ bias,  // [16384] fp32
    float* __restrict__ out)         // [64, 16384] fp32
{
    const int i    = blockIdx.x;          // row-block 0..127
    const int bt   = blockIdx.y;          // batch tile 0..3 (16 batches each)
    const int lane = threadIdx.x & 31;
    const int half = lane >> 4;           // 0: lanes 0-15, 1: lanes 16-31
    const int l16  = lane & 15;

    const int    rowbase = i * MC;        // i*128
    const float* brow    = bias + rowbase;

    // ---------------- bias-only region: output cols 0..79 ----------------
    // (weight rows 0..79 of the diagonal block are structurally zero)
    #pragma unroll 4
    for (int m = 0; m < 16; ++m) {
        float* orow = out + (size_t)(bt * 16 + m) * NM + rowbase;
        orow[lane]      = brow[lane];
        orow[lane + 32] = brow[lane + 32];
        if (lane < 16) orow[lane + 64] = brow[lane + 64];
    }

    // ---------------- WMMA band: output cols 80..127 ----------------
    // A fragment: x[bt*16 + m, i, 64..95]  (16x32, K split into 8 steps of 4)
    // A layout (16x4 f32): VGPR0 lanes0-15 K=+0, lanes16-31 K=+2; VGPR1 K=+1/+3.
    v2f a[8];
    {
        const float* xrow = x + (size_t)(bt * 16 + l16) * NM + rowbase + 64 + half * 2;
        #pragma unroll
        for (int s = 0; s < 8; ++s)
            a[s] = *(const v2f*)(xrow + s * 4);   // global_load_b64
    }

    // Output tiles t = 5,6,7 (cols 80..127). B[k][n] = W[rowbase+t*16+n, rowbase+64+k]
    // B layout (4x16 f32): VGPR0 lanes0-15 (K=+0,N=l16), lanes16-31 (K=+2,N=l16);
    //                      VGPR1 K=+1/+3. Consecutive-k pair -> one b64 load per lane.
    #pragma unroll
    for (int t = 5; t < 8; ++t) {
        const float bv = brow[t * 16 + l16];
        v8f acc;
        #pragma unroll
        for (int r = 0; r < 8; ++r) acc[r] = bv;   // C = broadcast bias row

        const float* wrow = W + (size_t)(rowbase + t * 16 + l16) * NM
                              + rowbase + 64 + half * 2;
        #pragma unroll
        for (int s = 0; s < 8; ++s) {
            v2f b = *(const v2f*)(wrow + s * 4);   // global_load_b64
            // 8 args: (neg_a, A, neg_b, B, c_mod, C, reuse_a, reuse_b)
            acc = __builtin_amdgcn_wmma_f32_16x16x4_f32(
                false, a[s], false, b, (short)0, acc, false, false);
        }

        // D layout: VGPR r -> M = r + 8*half, N = l16
        #pragma unroll
        for (int r = 0; r < 8; ++r) {
            out[(size_t)(bt * 16 + r + half * 8) * NM + rowbase + t * 16 + l16] = acc[r];
        }
    }
}

extern "C" void kernel_launch(void* const* d_in, const int* in_sizes, int n_in,
                              void* d_out, int out_size, void* d_ws, size_t ws_size,
                              hipStream_t stream) {
    const float* x    = (const float*)d_in[0];   // [64,128,128]
    const float* W    = (const float*)d_in[1];   // [16384,16384]
    const float* bias = (const float*)d_in[2];   // [16384]
    float*       out  = (float*)d_out;           // [64,128,128]

    dim3 grid(128, 4);   // (row-block i, batch tile)
    dim3 block(32);      // one wave32 per workgroup
    controlnet_band_wmma<<<grid, block, 0, stream>>>(x, W, bias, out);
}